// splineCNN_687194767571
// MI455X (gfx1250) — compile-verified
//
#include <hip/hip_runtime.h>
#include <hip/hip_bf16.h>
#include <math.h>

// ---------------------------------------------------------------------------
// SplineCNN forward on MI455X (gfx1250).
//
//  * per layer: pack Bcat = [W_0..W_19 | R] into WMMA-pair layout, then a
//    (FI,FO)-templated WMMA GEMM computes z[n, (K+1)*fo] = x @ Bcat with a
//    fully unrolled k-loop (V_WMMA_F32_16X16X4_F32, fp32 = reference dtype)
//    and compile-time strides (immediate-offset stores), then a float4 edge
//    pass interpolates z[src] and atomically scatters into agg[dst], then a
//    node pass applies agg/deg + z_R + b and ELU.
//  * N = 30000 = 1875*16 -> row dim exactly tile aligned.
//
// Workspace (floats): area[N] deg[N] frac[E] i0[E] xA[N*128] xB[N*128]
// agg[N*128] z[N*2688] Bp[128*2688]  -> ~372 MB (assumed available in d_ws).
// Accumulators re-zeroed every call -> deterministic.
// ---------------------------------------------------------------------------

typedef float v2f __attribute__((ext_vector_type(2)));
typedef float v8f __attribute__((ext_vector_type(8)));

#define NV   30000
#define NE   180000
#define NF   60000
#define KSP  20

// ---------------------------------------------------------------------------
__global__ void fzero_kernel(float* __restrict__ p, int n) {
    int t = blockIdx.x * blockDim.x + threadIdx.x;
    if (t < n) p[t] = 0.0f;
}

// ---------------------------------------------------------------------------
// per-face area scattered to all 3 vertices (|cross| is invariant under the
// three cyclic permutations used by the reference)
__global__ void face_area_kernel(const float* __restrict__ verts,
                                 const int* __restrict__ faces,
                                 float* __restrict__ area, int nf) {
    int f = blockIdx.x * blockDim.x + threadIdx.x;
    if (f >= nf) return;
    int ia = faces[f * 3 + 0], ib = faces[f * 3 + 1], ic = faces[f * 3 + 2];
    float ax = verts[ia * 3 + 0], ay = verts[ia * 3 + 1], az = verts[ia * 3 + 2];
    float bx = verts[ib * 3 + 0], by = verts[ib * 3 + 1], bz = verts[ib * 3 + 2];
    float cx = verts[ic * 3 + 0], cy = verts[ic * 3 + 1], cz = verts[ic * 3 + 2];
    float t1x = bx - ax, t1y = by - ay, t1z = bz - az;
    float t2x = cx - ax, t2y = cy - ay, t2z = cz - az;
    float crx = t1y * t2z - t1z * t2y;
    float cry = t1z * t2x - t1x * t2z;
    float crz = t1x * t2y - t1y * t2x;
    float ar = 0.5f * sqrtf(crx * crx + cry * cry + crz * crz);
    atomicAdd(area + ia, ar);
    atomicAdd(area + ib, ar);
    atomicAdd(area + ic, ar);
}

// ---------------------------------------------------------------------------
__global__ void edge_prep_kernel(const float* __restrict__ verts,
                                 const int* __restrict__ edges,
                                 int* __restrict__ ei0,
                                 float* __restrict__ efrac,
                                 float* __restrict__ deg, int ne) {
    int e = blockIdx.x * blockDim.x + threadIdx.x;
    if (e >= ne) return;
    int src = edges[e];
    int dst = edges[ne + e];
    float dx = verts[src * 3 + 0] - verts[dst * 3 + 0];
    float dy = verts[src * 3 + 1] - verts[dst * 3 + 1];
    float dz = verts[src * 3 + 2] - verts[dst * 3 + 2];
    float p = sqrtf(dx * dx + dy * dy + dz * dz);
    float v = fminf(fmaxf(p, 0.0f), 1.0f) * (float)(KSP - 1);
    float fl = fminf(fmaxf(floorf(v), 0.0f), (float)(KSP - 2));
    ei0[e] = (int)fl;
    efrac[e] = v - fl;
    atomicAdd(deg + dst, 1.0f);
}

// ---------------------------------------------------------------------------
__global__ void input_mlp_kernel(const float* __restrict__ verts,
                                 const float* __restrict__ area,
                                 const float* __restrict__ normals,
                                 const float* __restrict__ fw,
                                 const float* __restrict__ fb,
                                 float* __restrict__ x0, int n) {
    int t = blockIdx.x * blockDim.x + threadIdx.x;
    if (t >= n * 16) return;
    int node = t >> 4, o = t & 15;
    float f[7];
    f[0] = verts[node * 3 + 0];
    f[1] = verts[node * 3 + 1];
    f[2] = verts[node * 3 + 2];
    f[3] = area[node];
    f[4] = normals[node * 3 + 0];
    f[5] = normals[node * 3 + 1];
    f[6] = normals[node * 3 + 2];
    float acc = fb[o];
#pragma unroll
    for (int c = 0; c < 7; ++c) acc += f[c] * fw[o * 7 + c];
    x0[t] = acc;
}

// ---------------------------------------------------------------------------
// Pack Bcat[fi, Jpad] into WMMA-pair layout:
//   Bp[(f>>1)*Jpad*2 + j*2 + (f&1)] = Bcat[f, j]
// Bcat columns [0,K*fo) come from W[k][f][o], [K*fo, J) from R, rest zero.
__global__ void pack_b_kernel(const float* __restrict__ W,
                              const float* __restrict__ R,
                              float* __restrict__ Bp,
                              int fi, int fo, int fol2, int KF, int J, int Jpad) {
    int t = blockIdx.x * blockDim.x + threadIdx.x;
    if (t >= fi * Jpad) return;
    int f = t / Jpad;
    int j = t - f * Jpad;
    float v = 0.0f;
    if (j < KF) {
        int k = j >> fol2;
        int o = j & (fo - 1);
        v = W[((size_t)k * fi + f) * fo + o];
    } else if (j < J) {
        v = R[(size_t)f * fo + (j - KF)];
    }
    Bp[(size_t)(f >> 1) * Jpad * 2 + j * 2 + (f & 1)] = v;
}

// ---------------------------------------------------------------------------
// WMMA GEMM: Z[N, J] = X[N, FI] @ Bcat[FI, J], all strides compile-time.
// Block = 256 threads = 8 waves; one 16-row LDS strip x 8 column tiles.
// Inner loop per k-step: one ds_load_b64 (A pair) + one global_load_b64
// (packed B pair) + one V_WMMA_F32_16X16X4_F32; fully unrolled.
template <int FI, int FO>
__global__ __launch_bounds__(256)
void spline_gemm_wmma_kernel(const float* __restrict__ X,
                             const v2f* __restrict__ Bp,
                             float* __restrict__ Z) {
    constexpr int KF   = KSP * FO;          // spline columns
    constexpr int J    = KF + FO;           // + residual columns
    constexpr int JPAD = (J + 15) & ~15;

    __shared__ float As[16 * FI];

    const int wave = threadIdx.x >> 5;
    const int lane = threadIdx.x & 31;
    const int row0 = blockIdx.y * 16;
    const int col0 = (blockIdx.x * 8 + wave) * 16;

    // contiguous 16 x FI strip, float4 staging (rows consecutive in X)
    {
        const float4* X4 = (const float4*)(X + (size_t)row0 * FI);
        float4* As4 = (float4*)As;
#pragma unroll
        for (int idx = threadIdx.x; idx < 4 * FI; idx += 256)
            As4[idx] = X4[idx];
    }
    __syncthreads();

    if (col0 >= JPAD) return;  // wave-uniform: whole wave skips together

    // fragment coords (CDNA5 16x16x4 f32):
    //  A: lanes 0-15 -> M=lane, K=kk+{0,1}; lanes 16-31 -> M=lane-16, K=kk+{2,3}
    //  B: lanes 0-15 -> N=lane, K=kk+{0,1}; lanes 16-31 -> N=lane-16, K=kk+{2,3}
    const int rA   = lane & 15;
    const int koff = (lane >> 4) << 1;  // 0 or 2
    const int col  = col0 + (lane & 15);
    const v2f* As2 = (const v2f*)As;

    v8f acc = {0.f, 0.f, 0.f, 0.f, 0.f, 0.f, 0.f, 0.f};

#pragma unroll
    for (int kk = 0; kk < FI; kk += 4) {
        const int f = kk + koff;                       // even
        v2f a = As2[(rA * FI + f) >> 1];               // {A[m,f], A[m,f+1]}
        v2f b = Bp[(size_t)(f >> 1) * JPAD + col];     // {B[f,col], B[f+1,col]}
        acc = __builtin_amdgcn_wmma_f32_16x16x4_f32(
            false, a, false, b, (short)0, acc, false, false);
    }

    // C/D layout: VGPR v, lanes 0-15 -> M=v; lanes 16-31 -> M=v+8; N=lane&15
    if (col < J) {
        float* zp = Z + (size_t)(row0 + ((lane >> 4) << 3)) * J + col;
#pragma unroll
        for (int v = 0; v < 8; ++v)
            zp[(size_t)v * J] = acc[v];  // constant J -> immediate offsets
    }
}

// ---------------------------------------------------------------------------
// edge pass, 4 output channels per thread (fo multiple of 4):
// m = (1-frac)*z[src,i0,o..o+3] + frac*z[src,i0+1,o..o+3] -> atomic agg[dst]
__global__ void edge_agg4_kernel(const int* __restrict__ edges,
                                 const int* __restrict__ ei0,
                                 const float* __restrict__ efrac,
                                 const float* __restrict__ Z,
                                 float* __restrict__ agg,
                                 int ne, int fo, int ql2, int J) {
    int t = blockIdx.x * blockDim.x + threadIdx.x;
    const int q = fo >> 2;                 // float4 groups per edge
    if (t >= ne * q) return;
    int e = t >> ql2;
    int o = (t & (q - 1)) << 2;
    int src = edges[e];
    int dst = edges[ne + e];
    float fr = efrac[e];
    float w0 = 1.0f - fr;
    const float* zp = Z + (size_t)src * J + ei0[e] * fo + o;
    float4 z0 = *(const float4*)zp;
    float4 z1 = *(const float4*)(zp + fo);
    float* ap = agg + (size_t)dst * fo + o;
    atomicAdd(ap + 0, w0 * z0.x + fr * z1.x);
    atomicAdd(ap + 1, w0 * z0.y + fr * z1.y);
    atomicAdd(ap + 2, w0 * z0.z + fr * z1.z);
    atomicAdd(ap + 3, w0 * z0.w + fr * z1.w);
}

// scalar variant (fo == 1 layer)
__global__ void edge_agg1_kernel(const int* __restrict__ edges,
                                 const int* __restrict__ ei0,
                                 const float* __restrict__ efrac,
                                 const float* __restrict__ Z,
                                 float* __restrict__ agg,
                                 int ne, int J) {
    int e = blockIdx.x * blockDim.x + threadIdx.x;
    if (e >= ne) return;
    int src = edges[e];
    int dst = edges[ne + e];
    float fr = efrac[e];
    const float* zp = Z + (size_t)src * J + ei0[e];
    float m = (1.0f - fr) * zp[0] + fr * zp[1];
    atomicAdd(agg + dst, m);
}

// ---------------------------------------------------------------------------
// y = agg/deg + z_R + b, then ELU (except last layer)
__global__ void node_update_kernel(const float* __restrict__ agg,
                                   const float* __restrict__ deg,
                                   const float* __restrict__ Z,
                                   const float* __restrict__ bias,
                                   float* __restrict__ xout,
                                   int n, int fo, int fol2, int J, int KF,
                                   int apply_elu) {
    int t = blockIdx.x * blockDim.x + threadIdx.x;
    if (t >= n * fo) return;
    int node = t >> fol2;
    int o = t & (fo - 1);
    float d = fmaxf(deg[node], 1.0f);
    float y = agg[t] / d + Z[(size_t)node * J + KF + o] + bias[o];
    if (apply_elu) y = (y > 0.0f) ? y : expm1f(y);
    xout[t] = y;
}

// ---------------------------------------------------------------------------
template <int FI, int FO>
static inline void launch_gemm(const float* x, const float* Bp, float* z,
                               hipStream_t stream) {
    constexpr int J    = (KSP + 1) * FO;
    constexpr int JPAD = (J + 15) & ~15;
    dim3 grid(((JPAD / 16) + 7) / 8, NV / 16);
    spline_gemm_wmma_kernel<FI, FO><<<grid, 256, 0, stream>>>(
        x, (const v2f*)Bp, z);
}

extern "C" void kernel_launch(void* const* d_in, const int* in_sizes, int n_in,
                              void* d_out, int out_size, void* d_ws, size_t ws_size,
                              hipStream_t stream) {
    (void)in_sizes; (void)n_in; (void)out_size; (void)ws_size;

    const int fis[6]  = {16, 64, 128, 64, 32, 16};
    const int fos[6]  = {64, 128, 64, 32, 16, 1};
    const int fol2[6] = {6, 7, 6, 5, 4, 0};

    const float* verts   = (const float*)d_in[0];
    const int*   edges   = (const int*)d_in[1];
    const int*   faces   = (const int*)d_in[2];
    const float* normals = (const float*)d_in[3];
    const float* fc2w    = (const float*)d_in[4];
    const float* fc2b    = (const float*)d_in[5];
    const float* Wl[6]; const float* Rl[6]; const float* bl[6];
    for (int i = 0; i < 6; ++i) {
        Wl[i] = (const float*)d_in[6 + 3 * i];
        Rl[i] = (const float*)d_in[7 + 3 * i];
        bl[i] = (const float*)d_in[8 + 3 * i];
    }

    // workspace carve-up (all 4-byte elements; regions stay 16B aligned)
    float* ws    = (float*)d_ws;
    float* area  = ws;                          // N
    float* deg   = area + NV;                   // N
    float* efrac = deg + NV;                    // E
    int*   ei0   = (int*)(efrac + NE);          // E
    float* xA    = (float*)(ei0 + NE);          // N*128
    float* xB    = xA + (size_t)NV * 128;       // N*128
    float* agg   = xB + (size_t)NV * 128;       // N*128
    float* z     = agg + (size_t)NV * 128;      // N*2688 (max J)
    float* Bp    = z + (size_t)NV * 2688;       // 128*2688 packed operand

    const int TB = 256;

    // --- precompute -------------------------------------------------------
    fzero_kernel<<<(NV + TB - 1) / TB, TB, 0, stream>>>(area, NV);
    fzero_kernel<<<(NV + TB - 1) / TB, TB, 0, stream>>>(deg, NV);
    face_area_kernel<<<(NF + TB - 1) / TB, TB, 0, stream>>>(verts, faces, area, NF);
    edge_prep_kernel<<<(NE + TB - 1) / TB, TB, 0, stream>>>(verts, edges, ei0, efrac, deg, NE);
    input_mlp_kernel<<<(NV * 16 + TB - 1) / TB, TB, 0, stream>>>(
        verts, area, normals, fc2w, fc2b, xA, NV);

    // --- conv layers ------------------------------------------------------
    float* xcur = xA;
    float* xnext = xB;
    for (int L = 0; L < 6; ++L) {
        const int fi = fis[L], fo = fos[L], l2 = fol2[L];
        const int KF   = KSP * fo;              // spline columns
        const int J    = KF + fo;               // + residual columns
        const int Jpad = (J + 15) & ~15;

        fzero_kernel<<<(NV * fo + TB - 1) / TB, TB, 0, stream>>>(agg, NV * fo);

        pack_b_kernel<<<(fi * Jpad + TB - 1) / TB, TB, 0, stream>>>(
            Wl[L], Rl[L], Bp, fi, fo, l2, KF, J, Jpad);

        switch (L) {
        case 0: launch_gemm<16, 64>(xcur, Bp, z, stream); break;
        case 1: launch_gemm<64, 128>(xcur, Bp, z, stream); break;
        case 2: launch_gemm<128, 64>(xcur, Bp, z, stream); break;
        case 3: launch_gemm<64, 32>(xcur, Bp, z, stream); break;
        case 4: launch_gemm<32, 16>(xcur, Bp, z, stream); break;
        default: launch_gemm<16, 1>(xcur, Bp, z, stream); break;
        }

        if (fo >= 4) {
            const int q = fo >> 2;
            edge_agg4_kernel<<<(NE * q + TB - 1) / TB, TB, 0, stream>>>(
                edges, ei0, efrac, z, agg, NE, fo, l2 - 2, J);
        } else {
            edge_agg1_kernel<<<(NE + TB - 1) / TB, TB, 0, stream>>>(
                edges, ei0, efrac, z, agg, NE, J);
        }

        float* out = (L == 5) ? (float*)d_out : xnext;
        node_update_kernel<<<(NV * fo + TB - 1) / TB, TB, 0, stream>>>(
            agg, deg, z, bl[L], out, NV, fo, l2, J, KF, (L < 5) ? 1 : 0);

        float* tmp = xcur; xcur = xnext; xnext = tmp;
    }
}